// RNN_54314156425538
// MI455X (gfx1250) — compile-verified
//
#include <hip/hip_runtime.h>
#include <hip/hip_bf16.h>

typedef __attribute__((ext_vector_type(2))) float v2f;
typedef __attribute__((ext_vector_type(8))) float v8f;

#define T_STEPS 2048
#define BATCH   512
#define NIN     27
#define NROWS   (T_STEPS * BATCH)   // 1,048,576 (t,b) rows
#define NTILES  (NROWS / 16)        // 65,536 tiles of 16 rows

// ---------------------------------------------------------------------------
// Kernel 1: x_proj[row][g] = sum_k x[row][k] * W_ih[g][k] + b_ih[g] + b_hh[g]
// via V_WMMA_F32_16X16X4_F32 (fp32 in / fp32 acc), K padded 27 -> 28 (7x K=4),
// N padded 4 -> 16 (gates in columns 0..3).
// ---------------------------------------------------------------------------
__global__ void lstm_xproj_wmma(const float* __restrict__ x,
                                const float* __restrict__ W_ih,
                                const float* __restrict__ b_ih,
                                const float* __restrict__ b_hh,
                                float* __restrict__ xp)
{
    const int lane   = threadIdx.x & 31;
    const int wave   = (blockIdx.x * blockDim.x + threadIdx.x) >> 5;
    const int nwaves = (gridDim.x * blockDim.x) >> 5;

    const int mn    = lane & 15;            // A: row in tile; B/C/D: column (gate)
    const int khalf = (lane >> 4) << 1;     // K offset within a K=4 chunk: 0 or 2

    // B-matrix (K x N = 28 x 16; N<4 valid) resident in registers: 7 chunks.
    // Lane l holds B[k0][n], B[k0+1][n] with n = l&15, k0 = kk*4 + khalf.
    v2f bmat[7];
#pragma unroll
    for (int kk = 0; kk < 7; ++kk) {
        const int k0 = kk * 4 + khalf;
        float bx = 0.0f, by = 0.0f;
        if (mn < 4) {
            if (k0     < NIN) bx = W_ih[mn * NIN + k0];
            if (k0 + 1 < NIN) by = W_ih[mn * NIN + k0 + 1];
        }
        bmat[kk].x = bx;
        bmat[kk].y = by;
    }
    const float bias = (mn < 4) ? (b_ih[mn] + b_hh[mn]) : 0.0f;

    for (int tile = wave; tile < NTILES; tile += nwaves) {
        const float* __restrict__ arow = x + (size_t)(tile * 16 + mn) * NIN;

        v8f acc = {};
#pragma unroll
        for (int kk = 0; kk < 7; ++kk) {
            const int k0 = kk * 4 + khalf;
            v2f a;
            a.x = (k0     < NIN) ? arow[k0]     : 0.0f;  // only kk==6 can pad
            a.y = (k0 + 1 < NIN) ? arow[k0 + 1] : 0.0f;
            // 8 args: (neg_a, A, neg_b, B, c_mod, C, reuse_a, reuse_b)
            acc = __builtin_amdgcn_wmma_f32_16x16x4_f32(
                false, a, false, bmat[kk], (short)0, acc, false, false);
        }

        // D tile: lane holds column n = lane&15 for rows r + 8*(lane>=16).
        if (mn < 4) {
            const int rbase = tile * 16 + ((lane >> 4) << 3);
#pragma unroll
            for (int r = 0; r < 8; ++r) {
                xp[(size_t)(rbase + r) * 4 + mn] = acc[r] + bias;
            }
        }
    }
}

// ---------------------------------------------------------------------------
// Kernel 2: 512 independent scalar LSTM chains, one lane per batch element.
// Gate order i,f,g,o (PyTorch). Coalesced float4 gate loads per timestep.
// ---------------------------------------------------------------------------
__device__ __forceinline__ float fast_sigmoid(float v) {
    return 1.0f / (1.0f + __expf(-v));
}
__device__ __forceinline__ float fast_tanh(float v) {
    // tanh(x) = 1 - 2/(exp(2x)+1)
    return 1.0f - 2.0f / (__expf(2.0f * v) + 1.0f);
}

__global__ void lstm_scan(const float* __restrict__ xp,
                          const float* __restrict__ W_hh,
                          float* __restrict__ out)
{
    const int b = blockIdx.x * blockDim.x + threadIdx.x;
    if (b >= BATCH) return;

    const float wi = W_hh[0];
    const float wf = W_hh[1];
    const float wg = W_hh[2];
    const float wo = W_hh[3];

    float h = 0.0f, c = 0.0f;
    for (int t = 0; t < T_STEPS; ++t) {
        const float4 g = *(const float4*)(xp + ((size_t)t * BATCH + b) * 4);
        const float i  = fast_sigmoid(g.x + h * wi);
        const float f  = fast_sigmoid(g.y + h * wf);
        const float gg = fast_tanh   (g.z + h * wg);
        const float o  = fast_sigmoid(g.w + h * wo);
        c = f * c + i * gg;
        h = o * fast_tanh(c);
        out[(size_t)t * BATCH + b] = h;
    }
}

// ---------------------------------------------------------------------------
extern "C" void kernel_launch(void* const* d_in, const int* in_sizes, int n_in,
                              void* d_out, int out_size, void* d_ws, size_t ws_size,
                              hipStream_t stream)
{
    const float* x    = (const float*)d_in[0];  // (2048, 512, 27)
    const float* W_ih = (const float*)d_in[1];  // (4, 27)
    const float* W_hh = (const float*)d_in[2];  // (4, 1)
    const float* b_ih = (const float*)d_in[3];  // (4,)
    const float* b_hh = (const float*)d_in[4];  // (4,)
    float* out = (float*)d_out;                 // (2048, 512, 1)
    float* xp  = (float*)d_ws;                  // (T*B, 4) = 16 MB scratch

    // Phase 1: bandwidth-bound projection GEMM on the WMMA path.
    // 4096 waves -> 16 tiles/wave over 65,536 tiles.
    lstm_xproj_wmma<<<512, 256, 0, stream>>>(x, W_ih, b_ih, b_hh, xp);

    // Phase 2: latency-bound sequential scan, 512 independent lanes.
    lstm_scan<<<2, 256, 0, stream>>>(xp, W_hh, out);
}